// GDEC_67353677136180
// MI455X (gfx1250) — compile-verified
//
#include <hip/hip_runtime.h>
#include <math.h>

#define LRELU_NEG 0.01f

typedef __attribute__((ext_vector_type(2))) float v2f;
typedef __attribute__((ext_vector_type(4))) float v4f;
typedef __attribute__((ext_vector_type(8))) float v8f;

// Convert a generic pointer to a __shared__ object into the raw LDS byte
// offset needed by the async-DMA instructions (VDST = LDS address).
typedef __attribute__((address_space(3))) float lds_float_t;
__device__ __forceinline__ unsigned lds_addr_of(const float* p) {
    return (unsigned)(unsigned long long)(lds_float_t*)p;
}

// ---------------------------------------------------------------------------
// GCN preprocessing kernels
// ---------------------------------------------------------------------------

__global__ void k_deg(const int* __restrict__ edge, float* __restrict__ deg, int E) {
    int e = blockIdx.x * blockDim.x + threadIdx.x;
    if (e < E) atomicAdd(&deg[edge[E + e]], 1.0f);  // edge[1][e] = dst
}

__global__ void k_dinv(float* __restrict__ d, int N) {
    int i = blockIdx.x * blockDim.x + threadIdx.x;
    if (i < N) d[i] = rsqrtf(d[i] + 2.0f);          // improved: self-loop weight 2
}

// feat[n*B+b] = (x[b,n] + 0.1*noise[b,n]) * gcn_w   (transposed via LDS tile)
// agg [n*B+b] = 2*dinv[n]^2 * feat + gcn_b          (self-loop pre-accumulated)
__global__ void k_feat_agg(const float* __restrict__ x, const float* __restrict__ noise,
                           const float* __restrict__ dinv,
                           const float* __restrict__ gw, const float* __restrict__ gb,
                           float* __restrict__ feat, float* __restrict__ agg,
                           int B, int N) {
    __shared__ float tile[32][33];
    const int n0 = blockIdx.x * 32, b0 = blockIdx.y * 32;
    const float w = gw[0], bias = gb[0];
    #pragma unroll
    for (int r = 0; r < 32; r += 8) {
        size_t idx = (size_t)(b0 + threadIdx.y + r) * N + (n0 + threadIdx.x);
        tile[threadIdx.y + r][threadIdx.x] = (x[idx] + 0.1f * noise[idx]) * w;
    }
    __syncthreads();
    #pragma unroll
    for (int r = 0; r < 32; r += 8) {
        int n = n0 + threadIdx.y + r;
        int b = b0 + threadIdx.x;
        float f  = tile[threadIdx.x][threadIdx.y + r];
        float di = dinv[n];
        feat[(size_t)n * B + b] = f;
        agg [(size_t)n * B + b] = 2.0f * di * di * f + bias;
    }
}

// agg[dst, :] += feat[src, :] * dinv[src]*dinv[dst]   (coalesced atomics along B)
__global__ void k_scatter(const int* __restrict__ edge, const float* __restrict__ dinv,
                          const float* __restrict__ feat, float* __restrict__ agg,
                          int E, int B) {
    int e = blockIdx.x;
    int b = blockIdx.y * blockDim.x + threadIdx.x;
    int s = edge[e], d = edge[E + e];
    float nrm = dinv[s] * dinv[d];
    atomicAdd(&agg[(size_t)d * B + b], feat[(size_t)s * B + b] * nrm);
}

// gcn_out[b*N+n] = lrelu(agg[n*B+b])  (transpose back via LDS tile)
__global__ void k_gcn_out(const float* __restrict__ agg, float* __restrict__ out,
                          int B, int N) {
    __shared__ float tile[32][33];
    const int n0 = blockIdx.x * 32, b0 = blockIdx.y * 32;
    #pragma unroll
    for (int r = 0; r < 32; r += 8)
        tile[threadIdx.y + r][threadIdx.x] =
            agg[(size_t)(n0 + threadIdx.y + r) * B + (b0 + threadIdx.x)];
    __syncthreads();
    #pragma unroll
    for (int r = 0; r < 32; r += 8) {
        int b = b0 + threadIdx.y + r;
        int n = n0 + threadIdx.x;
        float v = tile[threadIdx.x][threadIdx.y + r];
        out[(size_t)b * N + n] = v >= 0.f ? v : LRELU_NEG * v;
    }
}

// ---------------------------------------------------------------------------
// LDS-staged fp32 WMMA GEMM with async double buffering:
//     C[M,Nout] = act(A[M,K] @ W[Nout,K]^T + bias)
//
// 256 threads = 8 waves. Block tile 128M x 64N, K chunk 32, LDS double-buffered.
//   - Full K-chunks are staged memory->LDS with GLOBAL_LOAD_ASYNC_TO_LDS_B128
//     (ASYNCcnt), issued for chunk i+1 right after the barrier so the DMA
//     overlaps the 32-WMMA burst on chunk i. Tail chunk (K%32) falls back to
//     a synchronous zero-filled stage (uniform branch).
//   - LDS row stride 36 floats: 16B-aligned b128 targets (36*4=9*16) and
//     bank-conflict-free 16-lane column reads (36r mod 64 = 4*(9r mod 16)).
//   - Out-of-range N rows of B clamp to row 0 (garbage only feeds columns
//     >= Nout, never stored); WMMA loop is branch-free, EXEC all-1s.
// ---------------------------------------------------------------------------
#define MBLK 128
#define NBLK 64
#define KC   32
#define LDSP 36

template <int ACT>   // 0=none 1=leaky_relu 2=sigmoid
__global__ __launch_bounds__(256) void
k_gemm_wmma(const float* __restrict__ A, const float* __restrict__ W,
            const float* __restrict__ bias, float* __restrict__ C,
            int M, int K, int Nout) {
    __shared__ __align__(16) float As[2][MBLK][LDSP];
    __shared__ __align__(16) float Bs[2][NBLK][LDSP];

    const int t     = threadIdx.x;
    const int wave  = t >> 5;
    const int lane  = t & 31;
    const int lm    = lane & 15;
    const int khalf = (lane >> 4) * 2;      // lanes 0-15: K 0/1, lanes 16-31: K 2/3
    const int m0    = blockIdx.x * MBLK;
    const int n0    = blockIdx.y * NBLK;

    // stage one K-chunk into buffer `buf` (async for full chunks, sync tail)
    auto stage = [&](int buf, int k0) {
        const bool full = (k0 + KC <= K);                  // uniform
        const float* abase = A + (size_t)m0 * K + k0;      // uniform SGPR base
        const float* wbase = W + k0;                       // uniform SGPR base
        #pragma unroll
        for (int i = 0; i < 4; ++i) {                      // A tile: 128x32
            int chunk = t + i * 256;
            int row = chunk >> 3;
            int c   = (chunk & 7) << 2;
            if (full) {
                unsigned dst  = lds_addr_of(&As[buf][row][c]);
                unsigned voff = (unsigned)(((size_t)row * K + c) * 4u);
                asm volatile("global_load_async_to_lds_b128 %0, %1, %2"
                             :: "v"(dst), "v"(voff), "s"(abase) : "memory");
            } else {
                v4f v = {0.f, 0.f, 0.f, 0.f};
                if (k0 + c + 4 <= K)
                    v = *(const v4f*)(abase + (size_t)row * K + c);
                As[buf][row][c + 0] = v.x; As[buf][row][c + 1] = v.y;
                As[buf][row][c + 2] = v.z; As[buf][row][c + 3] = v.w;
            }
        }
        #pragma unroll
        for (int i = 0; i < 2; ++i) {                      // B tile: 64x32
            int chunk = t + i * 256;
            int row = chunk >> 3;
            int c   = (chunk & 7) << 2;
            int n   = n0 + row;
            int nc  = (n < Nout) ? n : 0;                  // clamp (see header)
            if (full) {
                unsigned dst  = lds_addr_of(&Bs[buf][row][c]);
                unsigned voff = (unsigned)(((size_t)nc * K + c) * 4u);
                asm volatile("global_load_async_to_lds_b128 %0, %1, %2"
                             :: "v"(dst), "v"(voff), "s"(wbase) : "memory");
            } else {
                v4f v = {0.f, 0.f, 0.f, 0.f};
                if (k0 + c + 4 <= K)
                    v = *(const v4f*)(wbase + (size_t)nc * K + c);
                Bs[buf][row][c + 0] = v.x; Bs[buf][row][c + 1] = v.y;
                Bs[buf][row][c + 2] = v.z; Bs[buf][row][c + 3] = v.w;
            }
        }
    };

    v8f zero = {0.f, 0.f, 0.f, 0.f, 0.f, 0.f, 0.f, 0.f};
    v8f acc0 = zero, acc1 = zero, acc2 = zero, acc3 = zero;

    const int nChunks = (K + KC - 1) / KC;
    stage(0, 0);

    for (int ci = 0; ci < nChunks; ++ci) {
        const int cur = ci & 1;
        asm volatile("s_wait_asynccnt 0" ::: "memory");    // our DMA portion done
        __syncthreads();                                   // everyone's portion done
        if (ci + 1 < nChunks) stage(cur ^ 1, (ci + 1) * KC);  // overlap next DMA

        const float* arow = &As[cur][wave * 16 + lm][khalf];
        #pragma unroll
        for (int kk = 0; kk < KC; kk += 4) {
            v2f a  = *(const v2f*)(arow + kk);
            v2f b0 = *(const v2f*)&Bs[cur][ 0 + lm][kk + khalf];
            v2f b1 = *(const v2f*)&Bs[cur][16 + lm][kk + khalf];
            v2f b2 = *(const v2f*)&Bs[cur][32 + lm][kk + khalf];
            v2f b3 = *(const v2f*)&Bs[cur][48 + lm][kk + khalf];
#if __has_builtin(__builtin_amdgcn_wmma_f32_16x16x4_f32)
            acc0 = __builtin_amdgcn_wmma_f32_16x16x4_f32(false, a, false, b0, (short)0, acc0, false, false);
            acc1 = __builtin_amdgcn_wmma_f32_16x16x4_f32(false, a, false, b1, (short)0, acc1, false, false);
            acc2 = __builtin_amdgcn_wmma_f32_16x16x4_f32(false, a, false, b2, (short)0, acc2, false, false);
            acc3 = __builtin_amdgcn_wmma_f32_16x16x4_f32(false, a, false, b3, (short)0, acc3, false, false);
#else
            #pragma unroll
            for (int v = 0; v < 8; ++v) {   // host-pass / fallback only
                acc0[v] += a.x * b0.x + a.y * b0.y;
                acc1[v] += a.x * b1.x + a.y * b1.y;
                acc2[v] += a.x * b2.x + a.y * b2.y;
                acc3[v] += a.x * b3.x + a.y * b3.y;
            }
#endif
        }
        // next iteration's barrier protects buf `cur` from being overwritten
        // before all waves finished computing on it
    }

    // ---- epilogue: C/D layout VGPR v -> M = v (+8 for lanes 16-31), N = lane&15
    const int mrow = m0 + wave * 16 + ((lane >> 4) << 3);
#define EPILOGUE(J, ACCV)                                                     \
    {                                                                         \
        int n = n0 + (J) * 16 + lm;                                           \
        if (n < Nout) {                                                       \
            float bb = bias[n];                                               \
            _Pragma("unroll")                                                 \
            for (int v = 0; v < 8; ++v) {                                     \
                float val = ACCV[v] + bb;                                     \
                if (ACT == 1) val = val >= 0.f ? val : LRELU_NEG * val;       \
                if (ACT == 2) val = 1.f / (1.f + expf(-val));                 \
                C[(size_t)(mrow + v) * Nout + n] = val;                       \
            }                                                                 \
        }                                                                     \
    }
    EPILOGUE(0, acc0)
    EPILOGUE(1, acc1)
    EPILOGUE(2, acc2)
    EPILOGUE(3, acc3)
#undef EPILOGUE
}

// ---------------------------------------------------------------------------
// Discriminator head: out[b] = sigmoid(dot(H[b,:K], w) + b0), one wave per row
// ---------------------------------------------------------------------------
__global__ void k_disc_out(const float* __restrict__ H, const float* __restrict__ w,
                           const float* __restrict__ b0, float* __restrict__ out, int K) {
    int row = blockIdx.x, lane = threadIdx.x;
    const float* h = H + (size_t)row * K;
    float s = 0.f;
    for (int k = lane; k < K; k += 32) s += h[k] * w[k];
    #pragma unroll
    for (int off = 16; off > 0; off >>= 1) s += __shfl_down(s, off, 32);
    if (lane == 0) out[row] = 1.f / (1.f + expf(-(s + b0[0])));
}

// ---------------------------------------------------------------------------
// DEC Student-t soft assignment: q[b,k] with K=16 clusters, NZ=32 dims
// ---------------------------------------------------------------------------
__global__ void k_soft_assign(const float* __restrict__ z, const float* __restrict__ clusters,
                              float* __restrict__ q, int B) {
    __shared__ float cl[16 * 32];
    for (int i = threadIdx.x; i < 16 * 32; i += blockDim.x) cl[i] = clusters[i];
    __syncthreads();
    int b = blockIdx.x * blockDim.x + threadIdx.x;
    if (b >= B) return;
    float zz[32];
    #pragma unroll
    for (int d = 0; d < 32; ++d) zz[d] = z[(size_t)b * 32 + d];
    float qs[16], sum = 0.f;
    #pragma unroll
    for (int k = 0; k < 16; ++k) {
        float d2 = 0.f;
        #pragma unroll
        for (int d = 0; d < 32; ++d) { float df = zz[d] - cl[k * 32 + d]; d2 += df * df; }
        qs[k] = 1.f / (1.f + d2);
        sum += qs[k];
    }
    float inv = 1.f / sum;
    #pragma unroll
    for (int k = 0; k < 16; ++k) q[(size_t)b * 16 + k] = qs[k] * inv;
}

// ---------------------------------------------------------------------------
// Launch
// ---------------------------------------------------------------------------
extern "C" void kernel_launch(void* const* d_in, const int* in_sizes, int n_in,
                              void* d_out, int out_size, void* d_ws, size_t ws_size,
                              hipStream_t stream) {
    const int B = 2048, N = 4096, E = 65536;
    const int E1 = 500, E2 = 500, NZ = 32, H = 256;

    const float* x      = (const float*)d_in[0];
    const int*   edge   = (const int*)  d_in[1];
    const float* noise  = (const float*)d_in[2];
    const float* z_real = (const float*)d_in[3];
    const float* gw     = (const float*)d_in[4];
    const float* gb     = (const float*)d_in[5];
    const float* We1 = (const float*)d_in[6],  *be1 = (const float*)d_in[7];
    const float* We2 = (const float*)d_in[8],  *be2 = (const float*)d_in[9];
    const float* We3 = (const float*)d_in[10], *be3 = (const float*)d_in[11];
    const float* Wd1 = (const float*)d_in[12], *bd1 = (const float*)d_in[13];
    const float* Wd2 = (const float*)d_in[14], *bd2 = (const float*)d_in[15];
    const float* Wd3 = (const float*)d_in[16], *bd3 = (const float*)d_in[17];
    const float* Wq1 = (const float*)d_in[18], *bq1 = (const float*)d_in[19];
    const float* Wq2 = (const float*)d_in[20], *bq2 = (const float*)d_in[21];
    const float* Wq3 = (const float*)d_in[22], *bq3 = (const float*)d_in[23];
    const float* clusters = (const float*)d_in[24];

    // outputs (concatenated): x_bar [B*N], q [B*16], d_real [B], d_fake [B]
    float* out    = (float*)d_out;
    float* x_bar  = out;
    float* q_out  = out + (size_t)B * N;
    float* d_real = q_out + (size_t)B * 16;
    float* d_fake = d_real + B;

    // workspace layout (floats)
    float* ws    = (float*)d_ws;
    float* dinv  = ws;                                   // N
    float* feat  = dinv + N;                             // N*B  (reused as gcn_out)
    float* agg   = feat + (size_t)N * B;                 // N*B
    float* h1    = agg + (size_t)N * B;                  // B*500
    float* h2    = h1 + (size_t)B * E1;                  // B*500
    float* zlat  = h2 + (size_t)B * E2;                  // B*32
    float* dd1   = zlat + (size_t)B * NZ;                // B*500
    float* dd2   = dd1 + (size_t)B * 500;                // B*500
    float* hr1   = dd2 + (size_t)B * 500;                // B*256
    float* hr2   = hr1 + (size_t)B * H;                  // B*256
    float* hf1   = hr2 + (size_t)B * H;                  // B*256
    float* hf2   = hf1 + (size_t)B * H;                  // B*256

    // ---- GCN ----
    hipMemsetAsync(dinv, 0, (size_t)N * sizeof(float), stream);
    k_deg <<<(E + 255) / 256, 256, 0, stream>>>(edge, dinv, E);
    k_dinv<<<(N + 255) / 256, 256, 0, stream>>>(dinv, N);
    k_feat_agg<<<dim3(N / 32, B / 32), dim3(32, 8), 0, stream>>>(
        x, noise, dinv, gw, gb, feat, agg, B, N);
    k_scatter<<<dim3(E, B / 256), 256, 0, stream>>>(edge, dinv, feat, agg, E, B);
    k_gcn_out<<<dim3(N / 32, B / 32), dim3(32, 8), 0, stream>>>(agg, feat, B, N);
    float* gcn_out = feat;   // [B, N], leaky-relu applied

    // ---- encoder / decoder GEMM chain (fp32 WMMA) ----
    const int MG = B / MBLK;   // 16 blocks in M
    k_gemm_wmma<1><<<dim3(MG,  8), 256, 0, stream>>>(gcn_out, We1, be1, h1,   B, N,   E1);
    k_gemm_wmma<1><<<dim3(MG,  8), 256, 0, stream>>>(h1,      We2, be2, h2,   B, E1,  E2);
    k_gemm_wmma<1><<<dim3(MG,  1), 256, 0, stream>>>(h2,      We3, be3, zlat, B, E2,  NZ);
    k_gemm_wmma<1><<<dim3(MG,  8), 256, 0, stream>>>(zlat,    Wd1, bd1, dd1,  B, NZ,  500);
    k_gemm_wmma<1><<<dim3(MG,  8), 256, 0, stream>>>(dd1,     Wd2, bd2, dd2,  B, 500, 500);
    k_gemm_wmma<1><<<dim3(MG, 64), 256, 0, stream>>>(dd2,     Wd3, bd3, x_bar,B, 500, N);

    // ---- discriminator (real and fake) ----
    k_gemm_wmma<1><<<dim3(MG, 4), 256, 0, stream>>>(z_real, Wq1, bq1, hr1, B, NZ, H);
    k_gemm_wmma<1><<<dim3(MG, 4), 256, 0, stream>>>(hr1,    Wq2, bq2, hr2, B, H,  H);
    k_disc_out<<<B, 32, 0, stream>>>(hr2, Wq3, bq3, d_real, H);

    k_gemm_wmma<1><<<dim3(MG, 4), 256, 0, stream>>>(zlat, Wq1, bq1, hf1, B, NZ, H);
    k_gemm_wmma<1><<<dim3(MG, 4), 256, 0, stream>>>(hf1,  Wq2, bq2, hf2, B, H,  H);
    k_disc_out<<<B, 32, 0, stream>>>(hf2, Wq3, bq3, d_fake, H);

    // ---- DEC soft assignment ----
    k_soft_assign<<<(B + 255) / 256, 256, 0, stream>>>(zlat, clusters, q_out, B);

    (void)in_sizes; (void)n_in; (void)out_size; (void)ws_size;
}